// InputAverageModel_24790551233350
// MI455X (gfx1250) — compile-verified
//
#include <hip/hip_runtime.h>
#include <hip/hip_bf16.h>

// Problem constants (match reference)
#define B_  32
#define T_  36
#define N_  10000
#define F_  3
#define H_  10
#define R_  20
#define RPAD 32           // pad regions to 32 (two 16-wide WMMA column tiles)
#define K_BLOCKS 50       // K (=N) split across blocks for the WMMA GEMM
#define K_PER_BLOCK (N_ / K_BLOCKS)   // 200, multiple of 4

typedef __attribute__((ext_vector_type(2))) float v2f;
typedef __attribute__((ext_vector_type(8))) float v8f;

// ---------------------------------------------------------------------------
// 0) zero the accumulators (rsum[1024] + rcnt[1024] + g[2])
// ---------------------------------------------------------------------------
__global__ void zero_kernel(float* __restrict__ p, int n) {
    for (int i = threadIdx.x; i < n; i += blockDim.x) p[i] = 0.0f;
}

// ---------------------------------------------------------------------------
// 1) time_mean[b,n] = nanmean over T of seq[b,:,n,0].  0/0 -> NaN (all-missing)
//    Memory-bound: streams the full 138 MB of seq (stride-3 channel layout).
// ---------------------------------------------------------------------------
__global__ __launch_bounds__(256) void tm_kernel(const float* __restrict__ seq,
                                                 float* __restrict__ tm) {
    int idx = blockIdx.x * blockDim.x + threadIdx.x;
    if (idx >= B_ * N_) return;
    int b = idx / N_;
    int n = idx - b * N_;
    const float* p = seq + ((size_t)b * T_ * N_ + n) * F_;
    float s = 0.0f, cnt = 0.0f;
    #pragma unroll 4
    for (int t = 0; t < T_; ++t) {
        float x = p[(size_t)t * N_ * F_];
        bool v = (x == x);                 // !isnan
        s   += v ? x : 0.0f;
        cnt += v ? 1.0f : 0.0f;
    }
    tm[idx] = s / cnt;                     // NaN iff cnt==0 (matches reference)
}

// ---------------------------------------------------------------------------
// 2) Region segment-sum GEMM via V_WMMA_F32_16X16X4_F32.
//    rsum[b,r] = sum_n pz[b,n] * onehot[n,r]      (waves 0..3)
//    rcnt[b,r] = sum_n valid[b,n] * onehot[n,r]   (waves 4..7)
//    Each block handles a K-chunk of 200 n-values; partials via f32 atomics.
// ---------------------------------------------------------------------------
__global__ __launch_bounds__(256) void region_wmma_kernel(
        const float* __restrict__ tm, const int* __restrict__ cid,
        float* __restrict__ rsum, float* __restrict__ rcnt) {
    const int wave  = threadIdx.x >> 5;       // 0..7
    const int lane  = threadIdx.x & 31;
    const int mtile = wave & 1;               // b-tile: rows 0..15 / 16..31
    const int rtile = (wave >> 1) & 1;        // r-tile: cols 0..15 / 16..31
    const int mode  = wave >> 2;              // 0 = sum(pz), 1 = count(valid)

    const int l15   = lane & 15;
    const int khalf = lane >> 4;              // selects K pair {0,1} vs {2,3}
    const int arow  = mtile * 16 + l15;       // A-matrix row (b) for this lane
    const int rcol  = rtile * 16 + l15;       // B/D column (region) for this lane

    const float* tmrow = tm + (size_t)arow * N_;
    const int n_base = blockIdx.x * K_PER_BLOCK;

    v8f acc = {0.f, 0.f, 0.f, 0.f, 0.f, 0.f, 0.f, 0.f};

    for (int k0 = 0; k0 < K_PER_BLOCK; k0 += 4) {
        const int n = n_base + k0 + 2 * khalf;     // even -> 8B aligned

        // ---- A operand: two consecutive time_mean values, NaN-zeroed ----
        float2 t2 = *(const float2*)(tmrow + n);
        bool v0 = (t2.x == t2.x);
        bool v1 = (t2.y == t2.y);
        float pz0 = v0 ? t2.x : 0.0f, va0 = v0 ? 1.0f : 0.0f;
        float pz1 = v1 ? t2.y : 0.0f, va1 = v1 ? 1.0f : 0.0f;
        v2f a;
        a.x = mode ? va0 : pz0;                    // branch-free: EXEC stays ~0
        a.y = mode ? va1 : pz1;

        // ---- B operand: one-hot(cluster_id[n] == rcol) ----
        int2 c2 = *(const int2*)(cid + n);
        v2f bb;
        bb.x = (c2.x == rcol) ? 1.0f : 0.0f;
        bb.y = (c2.y == rcol) ? 1.0f : 0.0f;

        // D = A(16x4) * B(4x16) + C  -> v_wmma_f32_16x16x4_f32
        acc = __builtin_amdgcn_wmma_f32_16x16x4_f32(
                  /*neg_a=*/false, a, /*neg_b=*/false, bb,
                  /*c_mod=*/(short)0, acc, /*reuse_a=*/false, /*reuse_b=*/false);
    }

    // D layout: lane holds column rcol; VGPR i holds row (khalf*8 + i)
    float* dst = mode ? rcnt : rsum;
    #pragma unroll
    for (int i = 0; i < 8; ++i) {
        int brow = mtile * 16 + khalf * 8 + i;
        atomicAdd(&dst[brow * RPAD + rcol], acc[i]);
    }
}

// ---------------------------------------------------------------------------
// 3) Finalize: g1 = global nanmean(pred_speed) = sum(rsum)/sum(rcnt)
//              g2 = global nanmean(regional)   = mean of (rsum/rcnt | rcnt>0)
//    then write out_regional [B,10,R] with NaN -> g2.
// ---------------------------------------------------------------------------
__global__ __launch_bounds__(1024) void finalize_kernel(
        const float* __restrict__ rsum, const float* __restrict__ rcnt,
        float* __restrict__ g, float* __restrict__ out_regional) {
    __shared__ float s0[1024], s1[1024], s2[1024], s3[1024];
    __shared__ float g2s;
    const int t = threadIdx.x;                 // t = b*32 + r
    float sv = rsum[t];
    float cv = rcnt[t];
    bool  has = (cv > 0.0f);
    float ratio = has ? (sv / cv) : 0.0f;
    s0[t] = sv;
    s1[t] = cv;
    s2[t] = ratio;
    s3[t] = has ? 1.0f : 0.0f;
    __syncthreads();
    for (int off = 512; off > 0; off >>= 1) {
        if (t < off) {
            s0[t] += s0[t + off]; s1[t] += s1[t + off];
            s2[t] += s2[t + off]; s3[t] += s3[t + off];
        }
        __syncthreads();
    }
    if (t == 0) {
        float g1 = s0[0] / s1[0];              // NaN only if nothing valid
        float g2 = s2[0] / s3[0];
        g[0] = g1;
        g[1] = g2;
        g2s  = g2;
    }
    __syncthreads();
    const int b = t >> 5, r = t & 31;
    if (r < R_) {
        float v = has ? (sv / cv) : g2s;
        #pragma unroll
        for (int h = 0; h < H_; ++h)
            out_regional[((size_t)b * H_ + h) * R_ + r] = v;
    }
}

// ---------------------------------------------------------------------------
// 4) out_pred [B,10,N]: tile time_mean over HORIZON, NaN -> g1
// ---------------------------------------------------------------------------
__global__ __launch_bounds__(256) void pred_kernel(const float* __restrict__ tm,
                                                   const float* __restrict__ g,
                                                   float* __restrict__ out) {
    int idx = blockIdx.x * blockDim.x + threadIdx.x;
    if (idx >= B_ * N_) return;
    int b = idx / N_;
    int n = idx - b * N_;
    float v = tm[idx];
    if (!(v == v)) v = g[0];
    size_t base = (size_t)b * H_ * N_ + n;
    #pragma unroll
    for (int h = 0; h < H_; ++h)
        out[base + (size_t)h * N_] = v;
}

// ---------------------------------------------------------------------------
extern "C" void kernel_launch(void* const* d_in, const int* in_sizes, int n_in,
                              void* d_out, int out_size, void* d_ws, size_t ws_size,
                              hipStream_t stream) {
    const float* seq = (const float*)d_in[0];          // [B,T,N,F] f32
    const int*   cid = (const int*)d_in[1];            // [N] int
    float* out = (float*)d_out;                        // pred(3.2M) ++ regional(6400)
    float* ws  = (float*)d_ws;

    float* tm    = ws;                                 // B*N       = 320000 f
    float* rsum  = ws + (size_t)B_ * N_;               // 32*32     = 1024 f
    float* rcnt  = rsum + B_ * RPAD;                   // 1024 f
    float* g     = rcnt + B_ * RPAD;                   // 2 f
    float* out_regional = out + (size_t)B_ * H_ * N_;

    zero_kernel<<<1, 256, 0, stream>>>(rsum, 2 * B_ * RPAD + 2);
    tm_kernel<<<(B_ * N_ + 255) / 256, 256, 0, stream>>>(seq, tm);
    region_wmma_kernel<<<K_BLOCKS, 256, 0, stream>>>(tm, cid, rsum, rcnt);
    finalize_kernel<<<1, 1024, 0, stream>>>(rsum, rcnt, g, out_regional);
    pred_kernel<<<(B_ * N_ + 255) / 256, 256, 0, stream>>>(tm, g, out);
}